// HeteroDotPredictor_33028298506372
// MI455X (gfx1250) — compile-verified
//
#include <hip/hip_runtime.h>

typedef __attribute__((ext_vector_type(2))) float v2f;
typedef __attribute__((ext_vector_type(8))) float v8f;

#define D_FEAT 128

// One wave32 computes 16 edge scores per tile:
//   C(16x16) = U(16x128) * V(16x128)^T  accumulated via 32x V_WMMA_F32_16X16X4_F32,
//   scores = diag(C).
__global__ __launch_bounds__(256) void edge_dot_wmma_kernel(
    const float* __restrict__ h,
    const long long* __restrict__ src,
    const long long* __restrict__ dst,
    float* __restrict__ out,
    int n_edges)
{
    const int lane  = threadIdx.x & 31;
    const int wave  = (blockIdx.x * (blockDim.x >> 5)) + (threadIdx.x >> 5);
    const int nwave = (gridDim.x * blockDim.x) >> 5;

    const int m    = lane & 15;        // edge-in-tile this lane serves (A: M, B: N)
    const int koff = (lane >> 4) << 1; // K sub-offset: lanes 0-15 -> {0,1}, 16-31 -> {2,3}

    const int ntiles = (n_edges + 15) >> 4;

    for (int tile = wave; tile < ntiles; tile += nwave) {
        const int e0 = tile << 4;
        // Clamp gather index so EXEC stays all-1s through the WMMA loop.
        const int e  = (e0 + m < n_edges) ? (e0 + m) : (n_edges - 1);

        const long long si = src[e];
        const long long di = dst[e];
        const float* __restrict__ urow = h + si * (long long)D_FEAT;
        const float* __restrict__ vrow = h + di * (long long)D_FEAT;

        v8f c = {};
        #pragma unroll
        for (int k0 = 0; k0 < D_FEAT; k0 += 4) {
            // A[16x4] f32 layout: lane L holds A[M=L&15][K=koff..koff+1] in 2 VGPRs.
            v2f a = *(const v2f*)(urow + k0 + koff);
            // B[4x16] f32 layout is the symmetric striping: same per-lane addressing.
            v2f b = *(const v2f*)(vrow + k0 + koff);
            c = __builtin_amdgcn_wmma_f32_16x16x4_f32(
                    /*neg_a=*/false, a, /*neg_b=*/false, b,
                    /*c_mod=*/(short)0, c, /*reuse_a=*/false, /*reuse_b=*/false);
        }

        // Diagonal extraction (C/D 16x16 f32 layout):
        //   M=0..7  -> lanes 0..7,   VGPR idx = lane
        //   M=8..15 -> lanes 24..31, VGPR idx = lane - 24
        const bool lowHalf = (lane < 8);
        const bool hiHalf  = (lane >= 24);
        if (lowHalf || hiHalf) {
            const int r = lowHalf ? lane : (lane - 24);
            float s = c[0];
            s = (r == 1) ? c[1] : s;
            s = (r == 2) ? c[2] : s;
            s = (r == 3) ? c[3] : s;
            s = (r == 4) ? c[4] : s;
            s = (r == 5) ? c[5] : s;
            s = (r == 6) ? c[6] : s;
            s = (r == 7) ? c[7] : s;
            const int eo = e0 + (lowHalf ? lane : (lane - 16));
            if (eo < n_edges) out[eo] = s;
        }
    }
}

extern "C" void kernel_launch(void* const* d_in, const int* in_sizes, int n_in,
                              void* d_out, int out_size, void* d_ws, size_t ws_size,
                              hipStream_t stream) {
    const float*     h   = (const float*)d_in[0];
    const long long* src = (const long long*)d_in[1];  // int64 per reference
    const long long* dst = (const long long*)d_in[2];
    float*           out = (float*)d_out;

    const int n_edges = in_sizes[1];
    const int ntiles  = (n_edges + 15) >> 4;

    const int threads = 256;                 // 8 wave32 per block
    const int wpb     = threads >> 5;
    const int blocks  = (ntiles + wpb - 1) / wpb;

    edge_dot_wmma_kernel<<<blocks, threads, 0, stream>>>(h, src, dst, out, n_edges);
}